// ModelNew_25056839205038
// MI455X (gfx1250) — compile-verified
//
#include <hip/hip_runtime.h>
#include <cstdint>
#include <cstddef>

#define COLS        8192
#define THREADS     256
#define PER_THREAD  (COLS / THREADS)     // 32 floats per thread
#define CHUNK_STRIDE (PER_THREAD + 4)    // 36 floats: 16B-aligned, bank-skewed
#define NWAVES      (THREADS / 32)       // 8 waves (wave32)

typedef __attribute__((ext_vector_type(4))) float v4f;

__global__ __launch_bounds__(THREADS)
void cumprod_rows_kernel(const float* __restrict__ x, float* __restrict__ out) {
    __shared__ __align__(16) float tile[THREADS * CHUNK_STRIDE];  // ~36 KB
    __shared__ float wave_prod[NWAVES];

    const int t   = threadIdx.x;
    const int row = blockIdx.x;
    const float* row_in  = x   + (size_t)row * COLS;
    float*       row_out = out + (size_t)row * COLS;

    // ---- Stage row into LDS via CDNA5 async global->LDS copies (ASYNCcnt) ----
    // Coalesced on the global side: lane l of pass k reads float4 #(k*256 + l).
    // LDS destination is per-lane, written into the bank-skewed chunk layout.
    // Rows are single-touch and the 256 MiB stream exceeds the 192 MB L2, so
    // mark the loads non-temporal.
    #pragma unroll
    for (int k = 0; k < COLS / 4 / THREADS; ++k) {
        const int g     = k * THREADS + t;   // float4 index within the row
        const int chunk = g >> 3;            // owning 32-float chunk
        const int pos   = g & 7;             // float4 slot inside the chunk
        const unsigned lds_addr =
            (unsigned)(uintptr_t)&tile[chunk * CHUNK_STRIDE + pos * 4];
        const unsigned goff = (unsigned)g * 16u;   // byte offset within row
        asm volatile("global_load_async_to_lds_b128 %0, %1, %2 th:TH_LOAD_NT"
                     :
                     : "v"(lds_addr), "v"(goff), "s"(row_in)
                     : "memory");
    }
    asm volatile("s_wait_asynccnt 0x0" ::: "memory");
    __syncthreads();   // loads feed chunks owned by other waves

    // ---- Serial inclusive product over this thread's private chunk ----
    float vals[PER_THREAD];
    {
        const v4f* c4 = (const v4f*)&tile[t * CHUNK_STRIDE];
        float run = 1.0f;
        #pragma unroll
        for (int i = 0; i < PER_THREAD / 4; ++i) {
            v4f v = c4[i];
            run *= v.x; vals[4 * i + 0] = run;
            run *= v.y; vals[4 * i + 1] = run;
            run *= v.z; vals[4 * i + 2] = run;
            run *= v.w; vals[4 * i + 3] = run;
        }
    }
    const float chunk_total = vals[PER_THREAD - 1];

    // ---- wave32 inclusive scan (product) of chunk totals via shfl_up ----
    float incl = chunk_total;
    #pragma unroll
    for (int d = 1; d < 32; d <<= 1) {
        float up = __shfl_up(incl, (unsigned)d, 32);
        if ((t & 31) >= d) incl *= up;
    }
    float excl = __shfl_up(incl, 1u, 32);
    if ((t & 31) == 0) excl = 1.0f;

    const int wid = t >> 5;
    if ((t & 31) == 31) wave_prod[wid] = incl;   // wave total
    __syncthreads();

    // Exclusive product of preceding waves' totals (8 broadcast LDS reads).
    float wpre = 1.0f;
    #pragma unroll
    for (int w = 0; w < NWAVES; ++w)
        if (w < wid) wpre *= wave_prod[w];

    const float prefix = wpre * excl;

    // ---- Scale and store: contiguous 4 KB span per wave, non-temporal b128 ----
    v4f* o4 = (v4f*)(row_out + t * PER_THREAD);
    #pragma unroll
    for (int i = 0; i < PER_THREAD / 4; ++i) {
        v4f r;
        r.x = vals[4 * i + 0] * prefix;
        r.y = vals[4 * i + 1] * prefix;
        r.z = vals[4 * i + 2] * prefix;
        r.w = vals[4 * i + 3] * prefix;
        __builtin_nontemporal_store(r, &o4[i]);
    }
}

extern "C" void kernel_launch(void* const* d_in, const int* in_sizes, int n_in,
                              void* d_out, int out_size, void* d_ws, size_t ws_size,
                              hipStream_t stream) {
    const float* x = (const float*)d_in[0];
    float* out = (float*)d_out;
    const int rows = in_sizes[0] / COLS;   // 4096
    cumprod_rows_kernel<<<rows, THREADS, 0, stream>>>(x, out);
}